// _ProposalLayer_78297253806351
// MI455X (gfx1250) — compile-verified
//
#include <hip/hip_runtime.h>
#include <hip/hip_bf16.h>
#include <stdint.h>

// Problem constants (fixed by the reference).
#define NB        16
#define NA        12
#define NH        128
#define NW        128
#define NPROP     196608          // NA*NH*NW
#define PRE_NMS   6000
#define POST_NMS  300
#define NMS_TH    0.7f
#define TILE      4096            // keys per async tile (1024 thr * 4)
#define NTILES    48              // NPROP / TILE
#define CAND_CAP  8192

typedef uint32_t u32x4 __attribute__((ext_vector_type(4)));
typedef uint32_t u32x8 __attribute__((ext_vector_type(8)));

// ---------------------------------------------------------------------------
// CDNA5 async global->LDS DMA helpers (gfx1250, tracked by ASYNCcnt).
// ---------------------------------------------------------------------------
__device__ __forceinline__ uint32_t lds_addr(const void* p) {
  return (uint32_t)(uintptr_t)p;   // low 32 bits of flat addr == LDS offset
}
__device__ __forceinline__ void async_b128(uint32_t lds, const void* sbase, uint32_t voff) {
  asm volatile("global_load_async_to_lds_b128 %0, %1, %2"
               :: "v"(lds), "v"(voff), "s"(sbase) : "memory");
}
__device__ __forceinline__ void wait_async0() {
  asm volatile("s_wait_asynccnt 0" ::: "memory");
}
__device__ __forceinline__ void wait_async1() {
  asm volatile("s_wait_asynccnt 1" ::: "memory");
}

// ---------------------------------------------------------------------------
// Tensor Data Mover: 1-D contiguous tile, global -> LDS (tracked by
// TENSORcnt).  nelem8 = number of 8-byte elements.  D# per CDNA5 ISA:
//   group0: count=1 | lds_addr | global_addr[56:0] | type=2
//   group1: wg_mask=0, data_size=3 (8B), tensor_dim0=tile_dim0=nelem8,
//           tensor_dim1=1, tensor_dim0_stride=nelem8
// Issue from ONE wave only (TDM ignores EXEC; one issue per wave).
// ---------------------------------------------------------------------------
__device__ __forceinline__ void tdm_load_1d(uint32_t lds, const void* gptr, uint32_t nelem8) {
  const uint64_t ga = (uint64_t)(uintptr_t)gptr;
  u32x4 g0;
  g0.x = 0x00000001u;                                    // count=1, user D#
  g0.y = lds;                                            // lds_addr
  g0.z = (uint32_t)ga;                                   // global_addr[31:0]
  g0.w = ((uint32_t)(ga >> 32) & 0x01FFFFFFu) | 0x80000000u; // addr[56:32]|type=2
  u32x8 g1;
  g1.s0 = 0x00030000u;                                   // data_size=8B, mask=0
  g1.s1 = (nelem8 & 0xFFFFu) << 16;                      // tensor_dim0[15:0]
  g1.s2 = (nelem8 >> 16) | (1u << 16);                   // dim0[31:16] | dim1=1
  g1.s3 = (nelem8 & 0xFFFFu) << 16;                      // dim1[31:16]=0 | tile_dim0
  g1.s4 = 0u;                                            // tile_dim1=0, tile_dim2=0
  g1.s5 = nelem8;                                        // dim0_stride[31:0]
  g1.s6 = (nelem8 >> 16) & 0xFFFFu;                      // dim0_stride[47:32]
  g1.s7 = 0u;
  asm volatile("tensor_load_to_lds %0, %1" :: "s"(g0), "s"(g1) : "memory");
}

// ---------------------------------------------------------------------------
// K1: sortable keys from the objectness scores (upper NA channels).
// ---------------------------------------------------------------------------
__global__ void k_keys(const float* __restrict__ scores_map, uint32_t* __restrict__ keys) {
  const size_t gid = (size_t)blockIdx.x * 256 + threadIdx.x;
  if (gid >= (size_t)NB * NPROP) return;
  const size_t b   = gid / NPROP;
  const size_t rem = gid - b * NPROP;
  const float  s   = scores_map[b * (2u * NPROP) + NPROP + rem];
  uint32_t u = __float_as_uint(s);
  u = (u & 0x80000000u) ? ~u : (u | 0x80000000u);   // monotone float -> u32
  keys[gid] = u;
}

// ---------------------------------------------------------------------------
// K2: per-batch exact top-6000 threshold via 3-pass radix select
// (11+11+10 bits, 2048-bin LDS histograms), then compaction of keys >= T.
// All four streaming passes are fed by double-buffered async LDS DMA.
// ---------------------------------------------------------------------------
__global__ void __launch_bounds__(1024)
k_select_compact(const uint32_t* __restrict__ keys, uint64_t* __restrict__ cand,
                 int* __restrict__ counters) {
  const int b   = blockIdx.x;
  const int tid = threadIdx.x;
  const uint32_t* kb = keys + (size_t)b * NPROP;

  __shared__ uint32_t bins[2048];
  __shared__ uint32_t tiles[2][TILE];
  __shared__ uint32_t sh_prefix, sh_rank;
  __shared__ int      sh_cnt;

  if (tid == 0) { sh_prefix = 0u; sh_rank = PRE_NMS; }
  __syncthreads();

  const uint32_t l0 = lds_addr(&tiles[0][tid * 4]);
  const uint32_t l1 = lds_addr(&tiles[1][tid * 4]);
  const uint32_t pmask[3]  = {0x00000000u, 0xFFE00000u, 0xFFFFFC00u};
  const int      shifts[3] = {21, 10, 0};
  const int      nbinsA[3] = {2048, 2048, 1024};

  for (int pass = 0; pass < 3; ++pass) {
    for (int i = tid; i < 2048; i += 1024) bins[i] = 0u;
    __syncthreads();
    const uint32_t prefix = sh_prefix;
    const uint32_t pm = pmask[pass];
    const int      sh = shifts[pass];
    const uint32_t bm = (uint32_t)nbinsA[pass] - 1u;

    async_b128(l0, kb, (uint32_t)(tid * 16));              // prologue: tile 0
    for (int t = 0; t < NTILES; ++t) {
      const int cur = t & 1;
      if (t + 1 < NTILES) {
        async_b128(cur ? l0 : l1, kb, (uint32_t)((t + 1) * TILE * 4 + tid * 16));
        wait_async1();                                     // tile t landed
      } else {
        wait_async0();
      }
      __syncthreads();
#pragma unroll
      for (int k = 0; k < 4; ++k) {
        const uint32_t key = tiles[cur][tid * 4 + k];
        if ((key & pm) == prefix) atomicAdd(&bins[(key >> sh) & bm], 1u);
      }
      __syncthreads();
    }
    if (tid == 0) {
      uint32_t r = sh_rank, cum = 0u;
      for (int i = nbinsA[pass] - 1; i >= 0; --i) {
        const uint32_t c = bins[i];
        if (cum + c >= r) { sh_prefix = prefix | ((uint32_t)i << sh); sh_rank = r - cum; break; }
        cum += c;
      }
    }
    __syncthreads();
  }

  const uint32_t T = sh_prefix;            // exact 6000th-largest key
  if (tid == 0) sh_cnt = 0;
  __syncthreads();

  // Compaction pass: keys >= T, packed as (~key << 32) | flat_anchor_index so
  // the later ascending sort yields (score desc, index asc) == top_k order.
  async_b128(l0, kb, (uint32_t)(tid * 16));
  for (int t = 0; t < NTILES; ++t) {
    const int cur = t & 1;
    if (t + 1 < NTILES) {
      async_b128(cur ? l0 : l1, kb, (uint32_t)((t + 1) * TILE * 4 + tid * 16));
      wait_async1();
    } else {
      wait_async0();
    }
    __syncthreads();
#pragma unroll
    for (int k = 0; k < 4; ++k) {
      const uint32_t key = tiles[cur][tid * 4 + k];
      if (key >= T) {
        const int e = t * TILE + tid * 4 + k;       // e = c*16384 + p
        const int c = e >> 14, p = e & 16383;
        const uint32_t i_flat = (uint32_t)(p * NA + c);
        const int pos = atomicAdd(&sh_cnt, 1);
        if (pos < CAND_CAP)
          cand[(size_t)b * CAND_CAP + pos] = ((uint64_t)(~key) << 32) | (uint64_t)i_flat;
      }
    }
    __syncthreads();
  }
  if (tid == 0) counters[b] = sh_cnt < CAND_CAP ? sh_cnt : CAND_CAP;
}

// ---------------------------------------------------------------------------
// K3: bitonic sort of 8192 candidates in 64KB dynamic LDS, then decode+clip
// only the top-6000 boxes (gathered deltas/anchors).
// ---------------------------------------------------------------------------
__global__ void __launch_bounds__(1024)
k_sort_decode(const uint64_t* __restrict__ cand, const int* __restrict__ counters,
              const float* __restrict__ deltas, const float* __restrict__ anchors,
              const float* __restrict__ im_info, float4* __restrict__ boxes_ws) {
  extern __shared__ uint64_t sbuf[];
  const int b = blockIdx.x, tid = threadIdx.x;
  int cnt = counters[b]; if (cnt > CAND_CAP) cnt = CAND_CAP;
  const uint64_t* cb = cand + (size_t)b * CAND_CAP;
  for (int i = tid; i < CAND_CAP; i += 1024)
    sbuf[i] = (i < cnt) ? cb[i] : ~0ull;                   // pads sort last
  __syncthreads();

  for (unsigned k = 2; k <= CAND_CAP; k <<= 1)
    for (unsigned j = k >> 1; j > 0; j >>= 1) {
      for (unsigned i = tid; i < CAND_CAP; i += 1024) {
        const unsigned ixj = i ^ j;
        if (ixj > i) {
          const uint64_t a = sbuf[i], c = sbuf[ixj];
          const bool up = ((i & k) == 0);
          if ((a > c) == up) { sbuf[i] = c; sbuf[ixj] = a; }
        }
      }
      __syncthreads();
    }

  const float xmax = im_info[b * 3 + 1] - 1.0f;
  const float ymax = im_info[b * 3 + 0] - 1.0f;
  for (int t = tid; t < PRE_NMS; t += 1024) {
    const uint64_t pk = sbuf[t];
    float4 box = make_float4(0.f, 0.f, 0.f, 0.f);
    if (pk != ~0ull) {
      const uint32_t i_flat = (uint32_t)(pk & 0xFFFFFFFFu); // p*NA + a
      const int a = (int)(i_flat % (uint32_t)NA);
      const int p = (int)(i_flat / (uint32_t)NA);
      const float* an = anchors + (size_t)i_flat * 4;
      const float ax1 = an[0], ay1 = an[1], ax2 = an[2], ay2 = an[3];
      const float aw = ax2 - ax1 + 1.0f, ah = ay2 - ay1 + 1.0f;
      const float acx = ax1 + 0.5f * aw, acy = ay1 + 0.5f * ah;
      const size_t dbase = ((size_t)b * (4 * NA) + (size_t)a * 4) * (NH * NW) + (size_t)p;
      const float d0 = deltas[dbase];
      const float d1 = deltas[dbase + NH * NW];
      const float d2 = deltas[dbase + 2 * NH * NW];
      const float d3 = deltas[dbase + 3 * NH * NW];
      const float pcx = d0 * aw + acx, pcy = d1 * ah + acy;
      const float pw = __expf(d2) * aw, ph = __expf(d3) * ah;
      float x1 = pcx - 0.5f * pw, y1 = pcy - 0.5f * ph;
      float x2 = pcx + 0.5f * pw, y2 = pcy + 0.5f * ph;
      x1 = fminf(fmaxf(x1, 0.f), xmax); x2 = fminf(fmaxf(x2, 0.f), xmax);
      y1 = fminf(fmaxf(y1, 0.f), ymax); y2 = fminf(fmaxf(y2, 0.f), ymax);
      box = make_float4(x1, y1, x2, y2);
    }
    boxes_ws[(size_t)b * PRE_NMS + t] = box;
  }
}

// ---------------------------------------------------------------------------
// K4: greedy NMS entirely in LDS (~122KB dynamic, enabled by 320KB/WGP).
// The 96KB box tile is moved by ONE Tensor-Data-Mover DMA per workgroup.
// ---------------------------------------------------------------------------
__global__ void __launch_bounds__(1024)
k_nms_out(const float4* __restrict__ boxes_ws, float* __restrict__ out) {
  extern __shared__ char smem[];
  float4*       bx     = (float4*)smem;                     // 96000 B
  float*        area   = (float*)(smem + 96000);            // 24000 B
  unsigned int* valid  = (unsigned int*)(smem + 120000);    // 188 words
  int*          sel    = (int*)(smem + 120752);             // 300 ints
  int*          firstp = (int*)(smem + 121952);             // 1 int

  const int b = blockIdx.x, tid = threadIdx.x;
  const float4* src = boxes_ws + (size_t)b * PRE_NMS;

  if (tid < 32) {                                 // wave 0 only: one TDM issue
    tdm_load_1d(lds_addr(bx), src, PRE_NMS * 2u); // 12000 x 8B = 96000 B
    __builtin_amdgcn_s_wait_tensorcnt(0);
  }
  __syncthreads();

  for (int t = tid; t < PRE_NMS; t += 1024) {
    const float4 v = bx[t];
    area[t] = (v.z - v.x + 1.0f) * (v.w - v.y + 1.0f);
  }
  for (int w = tid; w < 188; w += 1024)
    valid[w] = (w < 187) ? 0xFFFFFFFFu : 0x0000FFFFu;       // 6000 bits set
  __syncthreads();

  for (int it = 0; it < POST_NMS; ++it) {
    if (tid == 0) *firstp = 0x7FFFFFFF;
    __syncthreads();
    for (int w = tid; w < 188; w += 1024) {
      const unsigned int v = valid[w];
      if (v) atomicMin(firstp, w * 32 + (__ffs(v) - 1));
    }
    __syncthreads();
    const int s = *firstp;
    if (tid == 0) sel[it] = (s == 0x7FFFFFFF) ? -1 : s;
    if (s != 0x7FFFFFFF) {
      const float4 sb = bx[s];
      const float  sa = area[s];
      for (int t = tid; t < PRE_NMS; t += 1024) {
        const unsigned int w = valid[t >> 5];
        if (w & (1u << (t & 31))) {
          const float4 ob = bx[t];
          const float iw = fminf(sb.z, ob.z) - fmaxf(sb.x, ob.x) + 1.0f;
          const float ih = fminf(sb.w, ob.w) - fmaxf(sb.y, ob.y) + 1.0f;
          const float inter = fmaxf(iw, 0.f) * fmaxf(ih, 0.f);
          const float iou = inter / (sa + area[t] - inter);
          if (iou > NMS_TH) atomicAnd(&valid[t >> 5], ~(1u << (t & 31)));
        }
      }
    }
    __syncthreads();
  }

  for (int t = tid; t < POST_NMS; t += 1024) {
    float* o = out + ((size_t)b * POST_NMS + t) * 5;
    const int s = sel[t];
    o[0] = (float)b;
    if (s >= 0) { const float4 v = bx[s]; o[1] = v.x; o[2] = v.y; o[3] = v.z; o[4] = v.w; }
    else        { o[1] = 0.f; o[2] = 0.f; o[3] = 0.f; o[4] = 0.f; }
  }
}

// ---------------------------------------------------------------------------
// Host launcher.
// ---------------------------------------------------------------------------
extern "C" void kernel_launch(void* const* d_in, const int* in_sizes, int n_in,
                              void* d_out, int out_size, void* d_ws, size_t ws_size,
                              hipStream_t stream) {
  (void)in_sizes; (void)n_in; (void)out_size; (void)ws_size;
  const float* scores  = (const float*)d_in[0];   // (16, 24, 128, 128)
  const float* deltas  = (const float*)d_in[1];   // (16, 48, 128, 128)
  const float* im_info = (const float*)d_in[2];   // (16, 3)
  const float* anchors = (const float*)d_in[3];   // (196608, 4)
  float* out = (float*)d_out;                     // (16, 300, 5)

  // Workspace layout (~15.2 MB): keys | cand | counters | boxes
  char* ws = (char*)d_ws;
  uint32_t* keys     = (uint32_t*)ws;                         // 12,582,912 B
  uint64_t* cand     = (uint64_t*)(ws + 12582912);            //  1,048,576 B
  int*      counters = (int*)     (ws + 13631488);            //         64 B
  float4*   boxes    = (float4*)  (ws + 13631552);            //  1,536,000 B

  const size_t nk = (size_t)NB * NPROP;
  k_keys<<<(unsigned)((nk + 255) / 256), 256, 0, stream>>>(scores, keys);
  k_select_compact<<<NB, 1024, 0, stream>>>(keys, cand, counters);
  k_sort_decode<<<NB, 1024, CAND_CAP * sizeof(uint64_t), stream>>>(
      cand, counters, deltas, anchors, im_info, boxes);
  k_nms_out<<<NB, 1024, 121984, stream>>>(boxes, out);
}